// MedianFilter_37709812859281
// MI455X (gfx1250) — compile-verified
//
#include <hip/hip_runtime.h>
#include <hip/hip_bf16.h>
#include <stdint.h>

// ---------------------------------------------------------------------------
// 5x5 median filter, reflect-101 padding, [B=8, C=3, H=512, W=512] fp32.
// Compute-bound: ~264 v_min/v_max per pixel (forgetful-selection network).
// Data staged per 16x16 tile into LDS via gfx1250 async global->LDS copies
// (ASYNCcnt / s_wait_asynccnt), falling back to plain loads if the builtin
// is unavailable on this toolchain.
// ---------------------------------------------------------------------------

#define TILE 16
#define HALO 2
#define LW   (TILE + 2 * HALO)   // 20
#define IMG  512

#if defined(__has_builtin)
#  if __has_builtin(__builtin_amdgcn_global_load_async_to_lds_b32)
#    define USE_ASYNC_LDS 1
#  endif
#endif

typedef __attribute__((address_space(1))) int* gas_i32_ptr;
typedef __attribute__((address_space(3))) int* lds_i32_ptr;

__device__ __forceinline__ void wait_async0() {
#if defined(__has_builtin) && __has_builtin(__builtin_amdgcn_s_wait_asynccnt)
    __builtin_amdgcn_s_wait_asynccnt(0);
#else
    asm volatile("s_wait_asynccnt 0" ::: "memory");
#endif
}

// compare-exchange: a=min, b=max (multiset-preserving)
__device__ __forceinline__ void s2(float& a, float& b) {
    float lo = fminf(a, b);
    float hi = fmaxf(a, b);
    a = lo;
    b = hi;
}

// Place min of w[0..N-1] at w[0], max at w[N-1]; w[1..N-2] keeps the rest
// (as a multiset). 3*floor(N/2) - 2 (+2 if odd) s2 units.
template <int N>
__device__ __forceinline__ void minmaxN(float* w) {
    constexpr int H = N / 2;
#pragma unroll
    for (int i = 0; i < H; ++i) s2(w[i], w[N - 1 - i]);
    constexpr int LOEND = (N & 1) ? H : H - 1;
#pragma unroll
    for (int i = 1; i <= LOEND; ++i) s2(w[0], w[i]);
#pragma unroll
    for (int i = H; i <= N - 2; ++i) s2(w[i], w[N - 1]);
}

__device__ __forceinline__ int reflect101(int q, int n) {
    q = (q < 0) ? -q : q;            // p=2 < n, one reflection suffices
    q = (q >= n) ? (2 * n - 2 - q) : q;
    return q;
}

__global__ __launch_bounds__(TILE * TILE)
void median5x5_kernel(const float* __restrict__ img, float* __restrict__ out) {
    __shared__ float tile[LW * LW];

    const int tx  = threadIdx.x;
    const int ty  = threadIdx.y;
    const int tid = ty * TILE + tx;
    const int c0  = blockIdx.x * TILE - HALO;   // leftmost input col of tile
    const int r0  = blockIdx.y * TILE - HALO;   // topmost input row of tile
    const float* __restrict__ chan = img + (size_t)blockIdx.z * (IMG * IMG);

    // ---- stage 20x20 halo tile into LDS (400 elems, 256 threads) ----------
#pragma unroll
    for (int pass = 0; pass < 2; ++pass) {
        const int e = tid + pass * (TILE * TILE);
        if (e < LW * LW) {
            const int lr = e / LW;
            const int lc = e - lr * LW;
            const int gr = reflect101(r0 + lr, IMG);
            const int gc = reflect101(c0 + lc, IMG);
            const float* src = chan + gr * IMG + gc;
#if defined(USE_ASYNC_LDS)
            __builtin_amdgcn_global_load_async_to_lds_b32(
                (gas_i32_ptr)(void*)const_cast<float*>(src),
                (lds_i32_ptr)&tile[e],
                /*offset=*/0, /*cpol=*/0);
#else
            tile[e] = *src;
#endif
        }
    }
#if defined(USE_ASYNC_LDS)
    wait_async0();
#endif
    __syncthreads();

    // ---- gather 5x5 window into VGPRs -------------------------------------
    float v[25];
#pragma unroll
    for (int i = 0; i < 5; ++i)
#pragma unroll
        for (int j = 0; j < 5; ++j)
            v[i * 5 + j] = tile[(ty + i) * LW + (tx + j)];

    // ---- exact median of 25 via forgetful selection -----------------------
    // Working set starts at rank+1 = 14; each round discards a provable
    // non-median min and max, then admits one new element. 132 s2 units.
    float w[14];
#pragma unroll
    for (int i = 0; i < 14; ++i) w[i] = v[i];
    minmaxN<14>(w);                 // survivors w[1..12]
    w[0] = v[14]; minmaxN<13>(w);
    w[0] = v[15]; minmaxN<12>(w);
    w[0] = v[16]; minmaxN<11>(w);
    w[0] = v[17]; minmaxN<10>(w);
    w[0] = v[18]; minmaxN<9>(w);
    w[0] = v[19]; minmaxN<8>(w);
    w[0] = v[20]; minmaxN<7>(w);
    w[0] = v[21]; minmaxN<6>(w);
    w[0] = v[22]; minmaxN<5>(w);
    w[0] = v[23]; minmaxN<4>(w);
    w[0] = v[24];                   // final 3 hold ranks {12,13,14}
    s2(w[0], w[1]); s2(w[1], w[2]); s2(w[0], w[1]);
    const float med = w[1];

    const int orow = blockIdx.y * TILE + ty;
    const int ocol = blockIdx.x * TILE + tx;
    out[(size_t)blockIdx.z * (IMG * IMG) + orow * IMG + ocol] = med;
}

// ---- mask passthrough: 128-bit vectorized copy ----------------------------
__global__ void copy_f32x4_kernel(const float4* __restrict__ src,
                                  float4* __restrict__ dst, int n4) {
    const int i = blockIdx.x * blockDim.x + threadIdx.x;
    if (i < n4) dst[i] = src[i];
}

extern "C" void kernel_launch(void* const* d_in, const int* in_sizes, int n_in,
                              void* d_out, int out_size, void* d_ws, size_t ws_size,
                              hipStream_t stream) {
    (void)n_in; (void)out_size; (void)d_ws; (void)ws_size;

    const float* img  = (const float*)d_in[0];   // [8,3,512,512] f32
    const float* mask = (const float*)d_in[1];   // [8,1,512,512] f32
    float* out = (float*)d_out;                  // concat(image_out, mask)

    const int BC = 8 * 3;                        // flattened batch*channel
    dim3 grid(IMG / TILE, IMG / TILE, BC);
    dim3 block(TILE, TILE, 1);
    median5x5_kernel<<<grid, block, 0, stream>>>(img, out);

    const int mask_n  = in_sizes[1];             // 2,097,152 floats
    const int n4      = mask_n / 4;
    float* mask_out   = out + (size_t)BC * IMG * IMG;  // 16B-aligned offset
    copy_f32x4_kernel<<<(n4 + 255) / 256, 256, 0, stream>>>(
        (const float4*)mask, (float4*)mask_out, n4);
}